// PyramidROIAlign_25580825215450
// MI455X (gfx1250) — compile-verified
//
#include <hip/hip_runtime.h>

// PyramidROIAlign for MI455X (gfx1250, wave32).
//
// Memory-bound gather (~12us HBM floor; fmaps fit in 192MB L2). Strategy:
//   phase 1: per (box, py) wave, y-lerp the 14 source columns into LDS
//            with fully coalesced 1KB NHWC row loads.
//   phase 2: x-pass as a real GEMM out[7x256] = Wx[7x14] @ T[14x256]
//            using V_WMMA_F32_16X16X4_F32 (4 K-steps, 16 channel tiles).

typedef __attribute__((ext_vector_type(2))) float v2f;
typedef __attribute__((ext_vector_type(4))) float f4;
typedef __attribute__((ext_vector_type(8))) float v8f;

#define PH 7
#define PW 7
#define CC 256
#define TSTRIDE 272  // 256 + 16 pad: half-wave LDS reads hit disjoint bank groups

__global__ __launch_bounds__(32)
void roi_align_wmma_kernel(const float* __restrict__ boxes,
                           const float* __restrict__ imgshape,
                           const float* __restrict__ P2,
                           const float* __restrict__ P3,
                           const float* __restrict__ P4,
                           const float* __restrict__ P5,
                           float* __restrict__ out,
                           int B, int N)
{
    __shared__ float T[16 * TSTRIDE];   // y-lerped columns, rows 14..15 zero pad

    const int box  = blockIdx.x;        // 0 .. B*N-1
    const int py   = blockIdx.y;        // 0 .. 6
    const int lane = threadIdx.x;       // 0 .. 31 (wave32)
    const int b    = box / N;

    const float y1 = boxes[box * 4 + 0];
    const float x1 = boxes[box * 4 + 1];
    const float y2 = boxes[box * 4 + 2];
    const float x2 = boxes[box * 4 + 3];

    // FPN level selection: lvl = clip(4 + round(log2(sqrt(h*w)*sqrt(area)/224)), 2, 5)
    const float area = imgshape[0] * imgshape[1];
    float lvlf = log2f(sqrtf((y2 - y1) * (x2 - x1)) * sqrtf(area) * (1.0f / 224.0f));
    lvlf = 4.0f + rintf(lvlf);                       // round-half-even, like jnp.round
    lvlf = fminf(5.0f, fmaxf(2.0f, lvlf));
    const int lvl = (int)lvlf;
    const int S   = 1024 >> lvl;                     // 256 / 128 / 64 / 32
    const float* fmap = (lvl == 2) ? P2 : (lvl == 3) ? P3 : (lvl == 4) ? P4 : P5;
    const size_t fbase = (size_t)b * S * S * CC;

    // y sample for this output row
    const float in_y = (y1 + (y2 - y1) * ((float)py / 6.0f)) * (float)(S - 1);
    const float y0f  = floorf(in_y);
    const float wy   = in_y - y0f;
    int y0 = (int)y0f;  y0 = min(max(y0, 0), S - 1);
    const int y1i = min(y0 + 1, S - 1);

    // ---- Phase 1: T[j][c] = lerp_y(fmap[y0, xcol(j), c], fmap[y1i, xcol(j), c])
    // columns: j = 2*px + e, xcol = (e ? x0+1 : x0), clamped. Lane owns 8 channels.
    const int cbase = lane * 8;
    for (int j = 0; j < 14; ++j) {
        const int   px   = j >> 1;
        const float in_x = (x1 + (x2 - x1) * ((float)px / 6.0f)) * (float)(S - 1);
        int x0 = (int)floorf(in_x);  x0 = min(max(x0, 0), S - 1);
        const int xcol = (j & 1) ? min(x0 + 1, S - 1) : x0;

        const float* top = fmap + fbase + ((size_t)y0  * S + xcol) * CC + cbase;
        const float* bot = fmap + fbase + ((size_t)y1i * S + xcol) * CC + cbase;
        f4 t0 = *(const f4*)(top);
        f4 t1 = *(const f4*)(top + 4);
        f4 b0 = *(const f4*)(bot);
        f4 b1 = *(const f4*)(bot + 4);
        f4 r0 = t0 + (b0 - t0) * wy;
        f4 r1 = t1 + (b1 - t1) * wy;
        *(f4*)(&T[j * TSTRIDE + cbase])     = r0;
        *(f4*)(&T[j * TSTRIDE + cbase + 4]) = r1;
    }
    // zero-pad K rows 14,15
    for (int j = 14; j < 16; ++j) {
        *(f4*)(&T[j * TSTRIDE + cbase])     = f4{0.0f, 0.0f, 0.0f, 0.0f};
        *(f4*)(&T[j * TSTRIDE + cbase + 4]) = f4{0.0f, 0.0f, 0.0f, 0.0f};
    }
    __syncthreads();  // single-wave WG: orders LDS stores vs loads

    // ---- Phase 2: out[px, c] = sum_j Wx[px][j] * T[j][c] via WMMA f32 16x16x4.
    // A layout (ISA 7.12.2): lanes 0-15 hold K = 0,1 ; lanes 16-31 hold K = 2,3.
    const int m     = lane & 15;             // output px (rows 7..15 are zero rows)
    const int kbase = (lane < 16) ? 0 : 2;

    // fractional x weight for output column m (only meaningful for m <= 6;
    // for m >= 7 the weight selectors never fire or hit zero-padded T rows)
    const float in_xm = (x1 + (x2 - x1) * ((float)m / 6.0f)) * (float)(S - 1);
    const float wxm   = in_xm - floorf(in_xm);

    const size_t obase = ((size_t)box * PH + py) * PW * CC;

    for (int ct = 0; ct < 16; ++ct) {
        v8f acc = {0.f, 0.f, 0.f, 0.f, 0.f, 0.f, 0.f, 0.f};
        const int c = ct * 16 + m;           // B/D column owned by this lane
        #pragma unroll
        for (int ks = 0; ks < 4; ++ks) {
            const int j0 = ks * 4 + kbase;
            // A[m][j0], A[m][j0+1]: sparse x-interp weights (2 nonzeros per row)
            v2f a;
            a.x = (j0     == 2 * m) ? (1.0f - wxm) : ((j0     == 2 * m + 1) ? wxm : 0.0f);
            a.y = (j0 + 1 == 2 * m) ? (1.0f - wxm) : ((j0 + 1 == 2 * m + 1) ? wxm : 0.0f);
            // B[j0][c], B[j0+1][c] from LDS (bank-conflict-free via TSTRIDE pad)
            v2f bm;
            bm.x = T[j0 * TSTRIDE + c];
            bm.y = T[(j0 + 1) * TSTRIDE + c];
            // D = A x B + C   (8 args: neg_a, A, neg_b, B, c_mod, C, reuse_a, reuse_b)
            acc = __builtin_amdgcn_wmma_f32_16x16x4_f32(false, a, false, bm,
                                                        (short)0, acc, false, false);
        }
        // D layout: lanes 0-15 hold N=lane, VGPR r = row M=r (rows 0..6 valid)
        if (lane < 16) {
            float* o = out + obase + ct * 16 + lane;
            #pragma unroll
            for (int r = 0; r < 7; ++r) o[(size_t)r * CC] = acc[r];
        }
    }
}

extern "C" void kernel_launch(void* const* d_in, const int* in_sizes, int n_in,
                              void* d_out, int out_size, void* d_ws, size_t ws_size,
                              hipStream_t stream) {
    const float* boxes    = (const float*)d_in[0];   // [B,N,4]
    const float* imgshape = (const float*)d_in[1];   // [2]
    const float* P2       = (const float*)d_in[2];   // [B,256,256,256]
    const float* P3       = (const float*)d_in[3];   // [B,128,128,256]
    const float* P4       = (const float*)d_in[4];   // [B, 64, 64,256]
    const float* P5       = (const float*)d_in[5];   // [B, 32, 32,256]
    float* out            = (float*)d_out;           // [B,N,7,7,256]

    const int BN = in_sizes[0] / 4;                  // B*N = 2000
    const int B  = 2;
    const int N  = BN / B;

    dim3 grid(BN, PH);
    roi_align_wmma_kernel<<<grid, 32, 0, stream>>>(boxes, imgshape, P2, P3, P4, P5,
                                                   out, B, N);
}